// A2M_32882269618302
// MI455X (gfx1250) — compile-verified
//
#include <hip/hip_runtime.h>
#include <hip/hip_bf16.h>
#include <cstdint>
#include <cstddef>

typedef __attribute__((ext_vector_type(2))) float v2f;
typedef __attribute__((ext_vector_type(8))) float v8f;
typedef int v4i __attribute__((vector_size(16)));

// ---------------------------------------------------------------------------
// concat [feat | turn | control | intersect] -> [Nmap, 132]
// ---------------------------------------------------------------------------
__global__ void cat_meta_kernel(const float* __restrict__ feat,
                                const float* __restrict__ turn,
                                const float* __restrict__ control,
                                const float* __restrict__ intersect,
                                float* __restrict__ out, int Nmap) {
    int gid = blockIdx.x * blockDim.x + threadIdx.x;
    int total = Nmap * 132;
    if (gid >= total) return;
    int r = gid / 132;
    int c = gid - r * 132;
    float v;
    if (c < 128)       v = feat[(size_t)r * 128 + c];
    else if (c < 130)  v = turn[(size_t)r * 2 + (c - 128)];
    else if (c == 130) v = control[r];
    else               v = intersect[r];
    out[gid] = v;
}

// ---------------------------------------------------------------------------
// edge distance MLP, K=2:  h1[e][j] = relu(dx*w[j][0] + dy*w[j][1] + b[j])
// ---------------------------------------------------------------------------
__global__ void edge_dist_kernel(const float* __restrict__ mc,
                                 const float* __restrict__ ac,
                                 const int* __restrict__ hi,
                                 const int* __restrict__ wi,
                                 const float* __restrict__ w1,
                                 const float* __restrict__ b1,
                                 float* __restrict__ out, int E) {
    int gid = blockIdx.x * blockDim.x + threadIdx.x;
    if (gid >= E * 128) return;
    int e = gid >> 7;
    int j = gid & 127;
    int h = hi[e], w = wi[e];
    float dx = mc[(size_t)h * 2 + 0] - ac[(size_t)w * 2 + 0];
    float dy = mc[(size_t)h * 2 + 1] - ac[(size_t)w * 2 + 1];
    float v = dx * w1[j * 2 + 0] + dy * w1[j * 2 + 1] + b1[j];
    out[gid] = fmaxf(v, 0.0f);
}

// ---------------------------------------------------------------------------
// helpers for CDNA5 async global->LDS weight staging
// (builtin wants: (v4i __device__* src, v4i __shared__* dst, imm off, imm cpol))
// ---------------------------------------------------------------------------
#define AS_GLOBAL_V4I(p) ((__attribute__((address_space(1))) v4i*)(uintptr_t)(p))
#define AS_LOCAL_V4I(p)  ((__attribute__((address_space(3))) v4i*)(uint32_t)(uintptr_t)(p))

__device__ __forceinline__ void wait_asynccnt0() {
#if __has_builtin(__builtin_amdgcn_s_wait_asynccnt)
    __builtin_amdgcn_s_wait_asynccnt(0);
#else
    asm volatile("s_wait_asynccnt 0" ::: "memory");
#endif
}

// ---------------------------------------------------------------------------
// Generic WMMA GEMM:  C[M,128] = sum_s  A_s[gather_s][:,0:K_s] @ W[:, k0_s:k0_s+K_s]^T
// W is row-major [128, Ktot] (torch Linear weight layout: out x in).
// Wave = 16-row x 128-col strip.  Block = 256 thr = 8 waves = 128 rows.
// If outIdx != null: scatter unsafeAtomicAdd into C[outIdx[row]] (index_add_).
// Weight segment staged in LDS (<= 128*132*4 = 67.5 KB of the 320 KB WGP LDS),
// via GLOBAL_LOAD_ASYNC_TO_LDS_B128 when the toolchain exposes it.
// ---------------------------------------------------------------------------
__global__ void __launch_bounds__(256)
gemm_seg_kernel(const float* A0, const int* g0, int K0,
                const float* A1, const int* g1, int K1,
                const float* A2, const int* g2, int K2,
                const float* __restrict__ W, int Ktot,
                float* __restrict__ C, const int* __restrict__ outIdx, int M) {
    extern __shared__ float ldsW[];
    const int tid  = threadIdx.x;
    const int wave = tid >> 5;
    const int lane = tid & 31;
    const int half = lane >> 4;   // 0: lanes 0-15, 1: lanes 16-31
    const int l16  = lane & 15;
    const int rowBase = blockIdx.x * 128 + wave * 16;

    v8f acc[8];
#pragma unroll
    for (int t = 0; t < 8; ++t) acc[t] = (v8f)0.0f;

    const float* segA[3] = {A0, A1, A2};
    const int*   segG[3] = {g0, g1, g2};
    const int    segK[3] = {K0, K1, K2};

    int k0 = 0;
    for (int s = 0; s < 3; ++s) {
        const int Ks = segK[s];
        if (Ks <= 0) break;                    // uniform across block
        const float* A = segA[s];
        const int*   G = segG[s];

        // stage W[:, k0:k0+Ks] into LDS as [128][Ks]
        __syncthreads();
        const int tot = 128 * Ks;              // multiple of 4; rows 16B-aligned
#if defined(__gfx1250__) && __has_builtin(__builtin_amdgcn_global_load_async_to_lds_b128)
        for (int i = tid * 4; i < tot; i += 256 * 4) {
            int n  = i / Ks;
            int kk = i - n * Ks;               // multiple of 4 -> chunk stays in row
            __builtin_amdgcn_global_load_async_to_lds_b128(
                AS_GLOBAL_V4I(W + (size_t)n * Ktot + k0 + kk),
                AS_LOCAL_V4I(ldsW + i), 0, 0);
        }
        wait_asynccnt0();
#else
        for (int i = tid; i < tot; i += 256) {
            int n  = i / Ks;
            int kk = i - n * Ks;
            ldsW[i] = W[(size_t)n * Ktot + k0 + kk];
        }
#endif
        __syncthreads();

        // A row for this lane (clamped; garbage rows masked at store time)
        int ar = rowBase + l16;
        if (ar > M - 1) ar = M - 1;
        const int arow = G ? G[ar] : ar;
        const float* Arow = A + (size_t)arow * Ks;
        __builtin_prefetch(Arow, 0, 0);        // global_prefetch_b8

        for (int kb = 0; kb < Ks; kb += 4) {
            // A frag: a[v] = A[m][kb + v + 2*half]  (8B global load)
            v2f a = *(const v2f*)(Arow + kb + 2 * half);
#pragma unroll
            for (int t = 0; t < 8; ++t) {
                const int n = t * 16 + l16;
                // B frag: b[v] = W[n][kb + v + 2*half]  (8B LDS load)
                v2f b = *(const v2f*)(&ldsW[n * Ks + kb + 2 * half]);
                acc[t] = __builtin_amdgcn_wmma_f32_16x16x4_f32(
                    false, a, false, b, (short)0, acc[t], false, false);
            }
        }
        k0 += Ks;
    }

    // C/D layout: acc[t][v] -> row (v + 8*half), col (t*16 + l16)
#pragma unroll
    for (int v = 0; v < 8; ++v) {
        const int r = rowBase + v + 8 * half;
        if (r < M) {
            if (outIdx) {
                const int orow = outIdx[r];
#pragma unroll
                for (int t = 0; t < 8; ++t)
                    unsafeAtomicAdd(&C[(size_t)orow * 128 + t * 16 + l16],
                                    acc[t][v]);
            } else {
#pragma unroll
                for (int t = 0; t < 8; ++t)
                    C[(size_t)r * 128 + t * 16 + l16] = acc[t][v];
            }
        }
    }
}

// ---------------------------------------------------------------------------
// GroupNorm(1 group over 128) + affine + (optional residual) + ReLU.
// One wave per row, 4 elems/lane, wave32 butterfly reductions.
// ---------------------------------------------------------------------------
__global__ void __launch_bounds__(256)
gn_relu_kernel(const float* __restrict__ in, float* __restrict__ out,
               const float* __restrict__ g, const float* __restrict__ b,
               const float* __restrict__ res, int M) {
    const int row  = blockIdx.x * 8 + (threadIdx.x >> 5);
    const int lane = threadIdx.x & 31;
    if (row >= M) return;
    const int c = lane * 4;
    const float4 v = *(const float4*)(in + (size_t)row * 128 + c);

    float s = v.x + v.y + v.z + v.w;
#pragma unroll
    for (int off = 16; off >= 1; off >>= 1) s += __shfl_xor(s, off, 32);
    const float mean = s * (1.0f / 128.0f);

    const float d0 = v.x - mean, d1 = v.y - mean, d2 = v.z - mean, d3 = v.w - mean;
    float q = d0 * d0 + d1 * d1 + d2 * d2 + d3 * d3;
#pragma unroll
    for (int off = 16; off >= 1; off >>= 1) q += __shfl_xor(q, off, 32);
    const float rinv = rsqrtf(q * (1.0f / 128.0f) + 1e-5f);

    const float4 gg = *(const float4*)(g + c);
    const float4 bb = *(const float4*)(b + c);
    float o0 = d0 * rinv * gg.x + bb.x;
    float o1 = d1 * rinv * gg.y + bb.y;
    float o2 = d2 * rinv * gg.z + bb.z;
    float o3 = d3 * rinv * gg.w + bb.w;
    if (res) {
        const float4 rr = *(const float4*)(res + (size_t)row * 128 + c);
        o0 += rr.x; o1 += rr.y; o2 += rr.z; o3 += rr.w;
    }
    float4 o;
    o.x = fmaxf(o0, 0.0f); o.y = fmaxf(o1, 0.0f);
    o.z = fmaxf(o2, 0.0f); o.w = fmaxf(o3, 0.0f);
    *(float4*)(out + (size_t)row * 128 + c) = o;
}

// ---------------------------------------------------------------------------
extern "C" void kernel_launch(void* const* d_in, const int* in_sizes, int n_in,
                              void* d_out, int out_size, void* d_ws, size_t ws_size,
                              hipStream_t stream) {
    const float* feat       = (const float*)d_in[0];
    const float* turn       = (const float*)d_in[1];
    const float* control    = (const float*)d_in[2];
    const float* intersect  = (const float*)d_in[3];
    const float* agents     = (const float*)d_in[4];
    const float* map_ctrs   = (const float*)d_in[5];
    const float* agent_ctrs = (const float*)d_in[6];
    const float* meta_w     = (const float*)d_in[7];
    const float* meta_g     = (const float*)d_in[8];
    const float* meta_b     = (const float*)d_in[9];
    const float* dist_w1    = (const float*)d_in[10];
    const float* dist_b1    = (const float*)d_in[11];
    const float* dist_w2    = (const float*)d_in[12];
    const float* dist_g2    = (const float*)d_in[13];
    const float* dist_b2    = (const float*)d_in[14];
    const float* query_w    = (const float*)d_in[15];
    const float* query_g    = (const float*)d_in[16];
    const float* query_b    = (const float*)d_in[17];
    const float* ctx_w1     = (const float*)d_in[18];
    const float* ctx_g1     = (const float*)d_in[19];
    const float* ctx_b1     = (const float*)d_in[20];
    const float* ctx_w2     = (const float*)d_in[21];
    const float* agt_w      = (const float*)d_in[22];
    const float* norm_g     = (const float*)d_in[23];
    const float* norm_b     = (const float*)d_in[24];
    const float* lin_w      = (const float*)d_in[25];
    const float* lin_g      = (const float*)d_in[26];
    const float* lin_b      = (const float*)d_in[27];
    const int*   hi         = (const int*)d_in[28];
    const int*   wi         = (const int*)d_in[29];

    const int Nmap = in_sizes[0] / 128;
    const int E    = in_sizes[28];

    float* ws = (float*)d_ws;
    size_t o = 0;
    float* catBuf = ws + o; o += (size_t)Nmap * 132;
    float* xtmp   = ws + o; o += (size_t)Nmap * 128;
    float* x1     = ws + o; o += (size_t)Nmap * 128;
    float* bufD   = ws + o; o += (size_t)E * 128;
    float* bufQ   = ws + o; o += (size_t)E * 128;
    float* bufC   = ws + o;
    float* x = (float*)d_out;   // x lives in d_out across the whole pipeline

    const dim3 blk(256);
    const int gemmGridN = (Nmap + 127) / 128;
    const int gemmGridE = (E + 127) / 128;
    const int gnGridN   = (Nmap + 7) / 8;
    const int gnGridE   = (E + 7) / 8;
    const size_t lds128 = 128 * 128 * sizeof(float);
    const size_t lds132 = 128 * 132 * sizeof(float);

    // ---- meta fuse: x = relu(GN(cat @ meta_w^T)) ----
    cat_meta_kernel<<<(Nmap * 132 + 255) / 256, blk, 0, stream>>>(
        feat, turn, control, intersect, catBuf, Nmap);
    gemm_seg_kernel<<<gemmGridN, blk, lds132, stream>>>(
        catBuf, nullptr, 132, nullptr, nullptr, 0, nullptr, nullptr, 0,
        meta_w, 132, xtmp, nullptr, Nmap);
    gn_relu_kernel<<<gnGridN, blk, 0, stream>>>(
        xtmp, x, meta_g, meta_b, nullptr, Nmap);

    for (int i = 0; i < 2; ++i) {
        const float* dw1 = dist_w1 + (size_t)i * 128 * 2;
        const float* db1 = dist_b1 + (size_t)i * 128;
        const float* dw2 = dist_w2 + (size_t)i * 128 * 128;
        const float* dg2 = dist_g2 + (size_t)i * 128;
        const float* db2 = dist_b2 + (size_t)i * 128;
        const float* qw  = query_w + (size_t)i * 128 * 128;
        const float* qg  = query_g + (size_t)i * 128;
        const float* qb  = query_b + (size_t)i * 128;
        const float* cw1 = ctx_w1  + (size_t)i * 128 * 384;
        const float* cg1 = ctx_g1  + (size_t)i * 128;
        const float* cb1 = ctx_b1  + (size_t)i * 128;
        const float* cw2 = ctx_w2  + (size_t)i * 128 * 128;
        const float* aw  = agt_w   + (size_t)i * 128 * 128;
        const float* ng  = norm_g  + (size_t)i * 128;
        const float* nb  = norm_b  + (size_t)i * 128;
        const float* lw  = lin_w   + (size_t)i * 128 * 128;
        const float* lg  = lin_g   + (size_t)i * 128;
        const float* lb  = lin_b   + (size_t)i * 128;

        if (E > 0) {
            // d = relu(GN(relu(d2 @ dist_w1^T + b1) @ dist_w2^T))
            edge_dist_kernel<<<(E * 128 + 255) / 256, blk, 0, stream>>>(
                map_ctrs, agent_ctrs, hi, wi, dw1, db1, bufD, E);
            gemm_seg_kernel<<<gemmGridE, blk, lds128, stream>>>(
                bufD, nullptr, 128, nullptr, nullptr, 0, nullptr, nullptr, 0,
                dw2, 128, bufD, nullptr, E);
            gn_relu_kernel<<<gnGridE, blk, 0, stream>>>(
                bufD, bufD, dg2, db2, nullptr, E);
            // q = relu(GN(x[hi] @ query_w^T))
            gemm_seg_kernel<<<gemmGridE, blk, lds128, stream>>>(
                x, hi, 128, nullptr, nullptr, 0, nullptr, nullptr, 0,
                qw, 128, bufQ, nullptr, E);
            gn_relu_kernel<<<gnGridE, blk, 0, stream>>>(
                bufQ, bufQ, qg, qb, nullptr, E);
            // c = relu(GN(cat(d,q,agents[wi]) @ ctx_w1^T))   (K = 3 x 128)
            gemm_seg_kernel<<<gemmGridE, blk, lds128, stream>>>(
                bufD, nullptr, 128, bufQ, nullptr, 128, agents, wi, 128,
                cw1, 384, bufC, nullptr, E);
            gn_relu_kernel<<<gnGridE, blk, 0, stream>>>(
                bufC, bufC, cg1, cb1, nullptr, E);
        }
        // xtmp = x @ agt_w^T ; then index_add_ of (c @ ctx_w2^T) at hi
        gemm_seg_kernel<<<gemmGridN, blk, lds128, stream>>>(
            x, nullptr, 128, nullptr, nullptr, 0, nullptr, nullptr, 0,
            aw, 128, xtmp, nullptr, Nmap);
        if (E > 0) {
            gemm_seg_kernel<<<gemmGridE, blk, lds128, stream>>>(
                bufC, nullptr, 128, nullptr, nullptr, 0, nullptr, nullptr, 0,
                cw2, 128, xtmp, hi, E);
        }
        // x1 = relu(GN(xtmp)); x = relu(GN(x1 @ lin_w^T) + res)
        gn_relu_kernel<<<gnGridN, blk, 0, stream>>>(
            xtmp, x1, ng, nb, nullptr, Nmap);
        gemm_seg_kernel<<<gemmGridN, blk, lds128, stream>>>(
            x1, nullptr, 128, nullptr, nullptr, 0, nullptr, nullptr, 0,
            lw, 128, x1, nullptr, Nmap);
        gn_relu_kernel<<<gnGridN, blk, 0, stream>>>(
            x1, x, lg, lb, /*res=*/x, Nmap);
    }
}